// GcnNet_58291296141744
// MI455X (gfx1250) — compile-verified
//
#include <hip/hip_runtime.h>
#include <hip/hip_bf16.h>

typedef __attribute__((ext_vector_type(2))) float v2f;
typedef __attribute__((ext_vector_type(8))) float v8f;

#define HID 96

// ---------------- small utility kernels ----------------

__global__ void k_fill(float* __restrict__ p, int n, float v) {
    int i = blockIdx.x * blockDim.x + threadIdx.x;
    if (i < n) p[i] = v;
}

__global__ void k_edge_deg(const int* __restrict__ dst, int E, float* __restrict__ deg) {
    int e = blockIdx.x * blockDim.x + threadIdx.x;
    if (e < E) atomicAdd(&deg[dst[e]], 1.0f);
}

__global__ void k_dinv(float* __restrict__ d, int n) {
    int i = blockIdx.x * blockDim.x + threadIdx.x;
    if (i < n) {
        float v = d[i];
        d[i] = (v > 0.0f) ? rsqrtf(v) : 0.0f;
    }
}

// ---------------- WMMA fp32 GEMM: out[M,N] = A[M,K] @ W[K,N] (+bias, relu) ----
// 256 threads = 8 waves; each wave computes one 16x16 tile via V_WMMA_F32_16X16X4_F32.
// blockIdx.y = N-tile; B slab (K x 16) staged in LDS.

__global__ __launch_bounds__(256)
void k_gemm_wmma(const float* __restrict__ A, const float* __restrict__ W,
                 const float* __restrict__ bias, float* __restrict__ out,
                 int M, int K, int N, int relu) {
    extern __shared__ float Bs[];            // K * 16 floats
    const int tid = threadIdx.x;
    const int nt  = blockIdx.y;

    for (int idx = tid; idx < K * 16; idx += 256) {
        int k = idx >> 4, n = idx & 15;
        Bs[idx] = W[k * N + nt * 16 + n];
    }
    __syncthreads();

    const int wave = tid >> 5;
    const int lane = tid & 31;
    const int mt   = blockIdx.x * 8 + wave;
    if (mt * 16 >= M) return;                // wave-uniform; EXEC stays all-1s

    const int mrow  = mt * 16 + (lane & 15); // A layout: lanes 0-15 rows, hi-half K+2
    const int khalf = (lane >> 4) * 2;       // 0 or 2
    const int ncol  = lane & 15;

    v8f acc = {};
    for (int k = 0; k < K; k += 4) {
        v2f a = *(const v2f*)(A + (size_t)mrow * K + k + khalf);   // A[m][k+khalf..+1]
        v2f b;
        const float* bp = Bs + (k + khalf) * 16 + ncol;            // B[k+khalf][n]
        b.x = bp[0];
        b.y = bp[16];
        acc = __builtin_amdgcn_wmma_f32_16x16x4_f32(
                  false, a, false, b, (short)0, acc, false, false);
    }

    const int orow0 = mt * 16 + (lane >> 4) * 8;   // C/D: VGPR r -> rows r and r+8
    const int ocol  = nt * 16 + ncol;
    float bv = bias ? bias[ocol] : 0.0f;
#pragma unroll
    for (int r = 0; r < 8; ++r) {
        float v = acc[r] + bv;
        if (relu) v = fmaxf(v, 0.0f);
        out[(size_t)(orow0 + r) * N + ocol] = v;
    }
}

// ---------------- GCN message passing ----------------

// agg[i][f] = xw[i][f] * dinv[i]^2   (self-loop term)
__global__ void k_self_init(const float* __restrict__ xw, const float* __restrict__ dinv,
                            float* __restrict__ agg) {
    int node = blockIdx.x, f = threadIdx.x;       // blockDim = 96
    float di = dinv[node];
    size_t i = (size_t)node * HID + f;
    agg[i] = xw[i] * di * di;
}

// one wave per edge: agg[dst] += xw[src] * dinv[src]*dinv[dst]
__global__ __launch_bounds__(256)
void k_edge_scatter(const int* __restrict__ src, const int* __restrict__ dst,
                    const float* __restrict__ dinv, const float* __restrict__ xw,
                    float* __restrict__ agg, int E) {
    int g    = blockIdx.x * blockDim.x + threadIdx.x;
    int e    = g >> 5;
    int lane = g & 31;
    if (e >= E) return;
    int s = src[e], d = dst[e];
    const float* xs = xw + (size_t)s * HID;
    float* ad = agg + (size_t)d * HID;
    __builtin_prefetch(xs + lane, 0, 0);
    float w = dinv[s] * dinv[d];
#pragma unroll
    for (int r = 0; r < 3; ++r) {                 // 96 = 3 * 32 lanes
        int f = lane + r * 32;
        atomicAdd(&ad[f], xs[f] * w);
    }
}

// in-place: agg = relu(agg + b)
__global__ void k_bias_relu(float* __restrict__ agg, const float* __restrict__ b) {
    int node = blockIdx.x, f = threadIdx.x;       // blockDim = 96
    size_t i = (size_t)node * HID + f;
    agg[i] = fmaxf(agg[i] + b[f], 0.0f);
}

// global add pool: g[batch[node]] += h[node]
__global__ void k_pool(const float* __restrict__ h, const int* __restrict__ batch,
                       float* __restrict__ g) {
    int node = blockIdx.x, f = threadIdx.x;       // blockDim = 96
    atomicAdd(&g[(size_t)batch[node] * HID + f], h[(size_t)node * HID + f]);
}

// out[g] = dot(hid[g][0:32], Wf2) + bf2  (one wave per graph, wave32 reduce)
__global__ void k_final(const float* __restrict__ hid, const float* __restrict__ Wf2,
                        const float* __restrict__ bf2, float* __restrict__ out, int G) {
    int t = blockIdx.x * blockDim.x + threadIdx.x;
    int g = t >> 5, lane = t & 31;
    if (g >= G) return;
    float v = hid[(size_t)g * 32 + lane] * Wf2[lane];
#pragma unroll
    for (int off = 16; off; off >>= 1) v += __shfl_xor(v, off, 32);
    if (lane == 0) out[g] = v + bf2[0];
}

// ---------------- launcher ----------------

extern "C" void kernel_launch(void* const* d_in, const int* in_sizes, int n_in,
                              void* d_out, int out_size, void* d_ws, size_t ws_size,
                              hipStream_t stream) {
    const float* x   = (const float*)d_in[0];
    const int*   ei  = (const int*)d_in[1];     // [2, E]
    const int*   bat = (const int*)d_in[2];
    const float* W1  = (const float*)d_in[3];  const float* b1  = (const float*)d_in[4];
    const float* W2  = (const float*)d_in[5];  const float* b2  = (const float*)d_in[6];
    const float* W3  = (const float*)d_in[7];  const float* b3  = (const float*)d_in[8];
    const float* W4  = (const float*)d_in[9];  const float* b4  = (const float*)d_in[10];
    const float* Wf1 = (const float*)d_in[11]; const float* bf1 = (const float*)d_in[12];
    const float* Wf2 = (const float*)d_in[13]; const float* bf2 = (const float*)d_in[14];

    const int N = in_sizes[0] / 32;   // 100000 (multiple of 16)
    const int E = in_sizes[1] / 2;    // 1600000
    const int G = out_size;           // 2048

    const int* srcp = ei;
    const int* dstp = ei + E;

    // workspace layout (floats)
    float* ws    = (float*)d_ws;
    float* dinv  = ws;
    float* xw    = dinv + N;
    float* agg   = xw  + (size_t)N * HID;
    float* gpool = agg + (size_t)N * HID;
    float* hid   = gpool + (size_t)G * HID;

    // degree (self-loop => init 1.0) -> dinv
    k_fill<<<(N + 255) / 256, 256, 0, stream>>>(dinv, N, 1.0f);
    k_edge_deg<<<(E + 255) / 256, 256, 0, stream>>>(dstp, E, dinv);
    k_dinv<<<(N + 255) / 256, 256, 0, stream>>>(dinv, N);

    const int mtiles = N / 16;                       // 6250
    dim3 gemm_grid((mtiles + 7) / 8, HID / 16);      // 782 x 6
    const int sblk = (E + 7) / 8;                    // one wave per edge

    // ---- layer 1 (K = 32) ----
    k_gemm_wmma<<<gemm_grid, 256, 32 * 16 * sizeof(float), stream>>>(
        x, W1, nullptr, xw, N, 32, HID, 0);
    k_self_init<<<N, HID, 0, stream>>>(xw, dinv, agg);
    k_edge_scatter<<<sblk, 256, 0, stream>>>(srcp, dstp, dinv, xw, agg, E);
    k_bias_relu<<<N, HID, 0, stream>>>(agg, b1);

    // ---- layers 2..4 (K = 96), h lives in agg, ping-pong through xw ----
    const float* Ws[3] = {W2, W3, W4};
    const float* bs[3] = {b2, b3, b4};
    for (int l = 0; l < 3; ++l) {
        k_gemm_wmma<<<gemm_grid, 256, HID * 16 * sizeof(float), stream>>>(
            agg, Ws[l], nullptr, xw, N, HID, HID, 0);
        k_self_init<<<N, HID, 0, stream>>>(xw, dinv, agg);
        k_edge_scatter<<<sblk, 256, 0, stream>>>(srcp, dstp, dinv, xw, agg, E);
        k_bias_relu<<<N, HID, 0, stream>>>(agg, bs[l]);
    }

    // ---- global add pool ----
    k_fill<<<(G * HID + 255) / 256, 256, 0, stream>>>(gpool, G * HID, 0.0f);
    k_pool<<<N, HID, 0, stream>>>(agg, bat, gpool);

    // ---- MLP head: relu(g @ Wf1 + bf1) via WMMA, then dot with Wf2 ----
    dim3 mlp_grid((G / 16 + 7) / 8, 32 / 16);        // 16 x 2
    k_gemm_wmma<<<mlp_grid, 256, HID * 16 * sizeof(float), stream>>>(
        gpool, Wf1, bf1, hid, G, HID, 32, 1);
    k_final<<<(G * 32 + 255) / 256, 256, 0, stream>>>(hid, Wf2, bf2, (float*)d_out, G);
}